// _VaDER_32384053412338
// MI455X (gfx1250) — compile-verified
//
#include <hip/hip_runtime.h>
#include <math.h>

// ---------------------------------------------------------------------------
// VaDER loss for MI455X (gfx1250).
// recon part: 402 MB read-once streaming -> bandwidth bound (~17us @ 23.3TB/s)
//             -> non-temporal loads so the stream does not thrash the 192MB L2.
// latent part: expanded quadratic forms as (8192x256)x(256x32) fp32 GEMMs
//              on V_WMMA_F32_16X16X4_F32 (exact fp32 products).
// ---------------------------------------------------------------------------

typedef __attribute__((ext_vector_type(2))) float v2f;
typedef __attribute__((ext_vector_type(4))) float v4f;
typedef __attribute__((ext_vector_type(8))) float v8f;

#define BN      8192
#define LN      256
#define KN      32
#define TD      4096.0f          /* N_STEPS * D_INPUT */
#define EPSF    1e-09f
#define LOG2PI  1.8378770664093453f

// workspace layout (float offsets)
#define OFF_D2   0               /* 2048 recon diff2 partials   */
#define OFF_MS   2048            /* 2048 recon mask partials    */
#define OFF_LAT  4096            /* 128 latent partials         */
#define OFF_IVZ  4224            /* 256x32  1/var      (pair-interleaved) */
#define OFF_MVZ  12416           /* 256x32  mu/var                        */
#define OFF_IVI  20608           /* 256x32  1/(var+eps)                   */
#define OFF_MVI  28800           /* 256x32  mu/(var+eps)                  */
#define OFF_CK   36992           /* 32 */
#define OFF_DK   37024           /* 32 */
#define OFF_LPHI 37056           /* 32 */

#define RECON_BLOCKS 2048
#define LAT_BLOCKS   128         /* 4 waves/block, 16 b-rows/wave -> 64 rows/block */

// ---------------------------------------------------------------------------
// Prep: per-cluster reciprocal matrices (transposed to [l][k], rows paired as
// float2 so a WMMA B-fragment is one 8-byte load) + per-cluster constants.
// ---------------------------------------------------------------------------
__global__ __launch_bounds__(256) void vader_prep(
    const float* __restrict__ mu_c, const float* __restrict__ var_c,
    const float* __restrict__ phi_c, float* __restrict__ ws)
{
    const int k = blockIdx.x;      // 0..31
    const int l = threadIdx.x;     // 0..255
    const float var = var_c[k * LN + l];
    const float mu  = mu_c [k * LN + l];
    const float ivz = 1.0f / var;
    const float ivi = 1.0f / (var + EPSF);
    // pair-interleaved transpose: element for (l,k) lives at ((l/2)*32+k)*2 + (l&1)
    const int p = (((l >> 1) * KN) + k) * 2 + (l & 1);
    ws[OFF_IVZ + p] = ivz;
    ws[OFF_MVZ + p] = mu * ivz;
    ws[OFF_IVI + p] = ivi;
    ws[OFF_MVI + p] = mu * ivi;

    const float lv = logf(var + EPSF);
    __shared__ float sc[256], sd[256];
    sc[l] = lv + LOG2PI + mu * mu * ivz;   // -> C_k
    sd[l] = lv + mu * mu * ivi;            // -> D_k
    __syncthreads();
    for (int s = 128; s > 0; s >>= 1) {
        if (l < s) { sc[l] += sc[l + s]; sd[l] += sd[l + s]; }
        __syncthreads();
    }
    if (l == 0) {
        ws[OFF_CK + k]   = sc[0];
        ws[OFF_DK + k]   = sd[0];
        ws[OFF_LPHI + k] = logf(phi_c[k] + EPSF);
    }
}

// ---------------------------------------------------------------------------
// Reconstruction: 16B grid-stride streaming reduction (bandwidth-bound).
// Non-temporal loads: 402 MB read exactly once -> do not displace L2 content.
// ---------------------------------------------------------------------------
__global__ __launch_bounds__(256) void vader_recon(
    const v4f* __restrict__ X, const v4f* __restrict__ M,
    const v4f* __restrict__ Xr, float* __restrict__ ws, int n4)
{
    float sd = 0.0f, sm = 0.0f;
    const int stride = gridDim.x * 256;
#pragma unroll 4
    for (int i = blockIdx.x * 256 + threadIdx.x; i < n4; i += stride) {
        const v4f x  = __builtin_nontemporal_load(&X[i]);
        const v4f r  = __builtin_nontemporal_load(&Xr[i]);
        const v4f mk = __builtin_nontemporal_load(&M[i]);
        const v4f d  = r - x;
        const v4f c  = d * d * mk;
        sd += c[0] + c[1] + c[2] + c[3];
        sm += mk[0] + mk[1] + mk[2] + mk[3];
    }
    __shared__ float s1[256], s2[256];
    s1[threadIdx.x] = sd; s2[threadIdx.x] = sm;
    __syncthreads();
    for (int s = 128; s > 0; s >>= 1) {
        if (threadIdx.x < s) { s1[threadIdx.x] += s1[threadIdx.x + s];
                               s2[threadIdx.x] += s2[threadIdx.x + s]; }
        __syncthreads();
    }
    if (threadIdx.x == 0) {
        ws[OFF_D2 + blockIdx.x] = s1[0];
        ws[OFF_MS + blockIdx.x] = s2[0];
    }
}

// ---------------------------------------------------------------------------
// Latent: WMMA f32 16x16x4. One wave = 16 b-rows; 2 k-tiles cover K=32.
// Per K-step (4 of L): 8 independent v_wmma_f32_16x16x4_f32.
// A-side data (z, mu_tilde, stddev_tilde) is read exactly once -> NT loads.
// B-side matrices are reused by all 512 waves -> default policy (L2 resident).
// ---------------------------------------------------------------------------
__global__ __launch_bounds__(128) void vader_latent(
    const float* __restrict__ z, const float* __restrict__ mu_t,
    const float* __restrict__ st_t, float* __restrict__ ws)
{
    const int lane = threadIdx.x & 31;
    const int w    = threadIdx.x >> 5;        // wave in block (0..3)
    const int n    = lane & 15;               // A-row m / B,C col n
    const int half = lane >> 4;               // 0: K=0,1 ; 1: K=2,3
    const int k0   = half << 1;
    const int b0   = blockIdx.x * 64 + w * 16;
    const int brow = b0 + n;

    const v2f* __restrict__ Zr = (const v2f*)(z    + (size_t)brow * LN);
    const v2f* __restrict__ Mr = (const v2f*)(mu_t + (size_t)brow * LN);
    const v2f* __restrict__ Sr = (const v2f*)(st_t + (size_t)brow * LN);

    const v2f* __restrict__ IVZ = (const v2f*)(ws + OFF_IVZ);
    const v2f* __restrict__ MVZ = (const v2f*)(ws + OFF_MVZ);
    const v2f* __restrict__ IVI = (const v2f*)(ws + OFF_IVI);
    const v2f* __restrict__ MVI = (const v2f*)(ws + OFF_MVI);

    v8f zero;
#pragma unroll
    for (int i = 0; i < 8; ++i) zero[i] = 0.0f;
    v8f qz[2] = { zero, zero };   // Qz accumulators, k-tiles 0/1
    v8f qi[2] = { zero, zero };   // Qi accumulators
    float p3 = 0.0f;              // running sum of stddev_tilde (for loss3)

    for (int l0 = 0; l0 < LN; l0 += 4) {
        const int la = l0 + k0;               // even; this lane's first K col
        const v2f zf = __builtin_nontemporal_load(&Zr[la >> 1]);
        const v2f mf = __builtin_nontemporal_load(&Mr[la >> 1]);
        const v2f sf = __builtin_nontemporal_load(&Sr[la >> 1]);
        p3 += sf[0] + sf[1];

        v2f az2; az2[0] = zf[0] * zf[0];             az2[1] = zf[1] * zf[1];
        v2f azm; azm[0] = -2.0f * zf[0];             azm[1] = -2.0f * zf[1];
        v2f ai1; ai1[0] = expf(sf[0]) + mf[0]*mf[0]; ai1[1] = expf(sf[1]) + mf[1]*mf[1];
        v2f ai2; ai2[0] = -2.0f * mf[0];             ai2[1] = -2.0f * mf[1];

        const int pbase = (la >> 1) * KN;         // float2-element base row
#pragma unroll
        for (int kt = 0; kt < 2; ++kt) {
            const int idx = pbase + kt * 16 + n;
            const v2f bz1 = IVZ[idx];
            const v2f bz2 = MVZ[idx];
            const v2f bq1 = IVI[idx];
            const v2f bq2 = MVI[idx];
            qz[kt] = __builtin_amdgcn_wmma_f32_16x16x4_f32(
                false, az2, false, bz1, (short)0, qz[kt], false, false);
            qz[kt] = __builtin_amdgcn_wmma_f32_16x16x4_f32(
                false, azm, false, bz2, (short)0, qz[kt], false, false);
            qi[kt] = __builtin_amdgcn_wmma_f32_16x16x4_f32(
                false, ai1, false, bq1, (short)0, qi[kt], false, false);
            qi[kt] = __builtin_amdgcn_wmma_f32_16x16x4_f32(
                false, ai2, false, bq2, (short)0, qi[kt], false, false);
        }
    }

    // per-lane cluster constants for its two k columns (n and n+16)
    const float ckA = ws[OFF_CK + n],        ckB = ws[OFF_CK + 16 + n];
    const float dkA = ws[OFF_DK + n],        dkB = ws[OFF_DK + 16 + n];
    const float lpA = ws[OFF_LPHI + n],      lpB = ws[OFF_LPHI + 16 + n];

    __shared__ float Plog[4][16][33];   // padded: conflict-free row reads
    __shared__ float Pin [4][16][33];
    __shared__ float rl  [4][16];
    __shared__ float p3s [128];
    __shared__ float lphi_s[32];
    if (threadIdx.x < 32) lphi_s[threadIdx.x] = ws[OFF_LPHI + threadIdx.x];

    // C/D fragment layout: lane holds (M = r + 8*half, N = n), VGPR r = 0..7
#pragma unroll
    for (int r = 0; r < 8; ++r) {
        const int mm = r + (half << 3);
        Plog[w][mm][n]      = lpA - 0.5f * (ckA + qz[0][r]);
        Plog[w][mm][16 + n] = lpB - 0.5f * (ckB + qz[1][r]);
        Pin [w][mm][n]      = dkA + qi[0][r];
        Pin [w][mm][16 + n] = dkB + qi[1][r];
    }
    p3s[threadIdx.x] = p3;
    __syncthreads();

    if (lane < 16) {                       // one lane per b-row: softmax over K
        const int mm = lane;
        float mx = -1e30f;
        for (int k = 0; k < KN; ++k) mx = fmaxf(mx, Plog[w][mm][k]);
        float s = 0.0f;
        for (int k = 0; k < KN; ++k) s += expf(Plog[w][mm][k] - mx);
        const float lse = mx + logf(s);
        float a1 = 0.0f, a2 = 0.0f;
        for (int k = 0; k < KN; ++k) {
            const float lg = Plog[w][mm][k] - lse;   // log gamma
            const float g  = expf(lg);
            a1 += g * Pin[w][mm][k];
            a2 += g * (lphi_s[k] - lg);
        }
        rl[w][mm] = 0.5f * a1 - a2;        // loss1 + loss2 for this sample
    }
    __syncthreads();

    if (threadIdx.x == 0) {                // fixed-order -> deterministic
        float tot = 0.0f;
        for (int ww = 0; ww < 4; ++ww)
            for (int i = 0; i < 16; ++i) tot += rl[ww][i];
        float sp3 = 0.0f;
        for (int t = 0; t < 128; ++t) sp3 += p3s[t];
        tot += -0.5f * (64.0f * (float)LN + sp3);   // loss3 over 64 samples
        ws[OFF_LAT + blockIdx.x] = tot;
    }
}

// ---------------------------------------------------------------------------
// Final deterministic reduction + scalar assembly.
// ---------------------------------------------------------------------------
__global__ __launch_bounds__(256) void vader_final(
    const float* __restrict__ ws, float* __restrict__ out)
{
    __shared__ float s1[256], s2[256], s3[256];
    float a = 0.0f, b = 0.0f, c = 0.0f;
    for (int i = threadIdx.x; i < RECON_BLOCKS; i += 256) {
        a += ws[OFF_D2 + i];
        b += ws[OFF_MS + i];
    }
    for (int i = threadIdx.x; i < LAT_BLOCKS; i += 256) c += ws[OFF_LAT + i];
    s1[threadIdx.x] = a; s2[threadIdx.x] = b; s3[threadIdx.x] = c;
    __syncthreads();
    for (int s = 128; s > 0; s >>= 1) {
        if (threadIdx.x < s) {
            s1[threadIdx.x] += s1[threadIdx.x + s];
            s2[threadIdx.x] += s2[threadIdx.x + s];
            s3[threadIdx.x] += s3[threadIdx.x + s];
        }
        __syncthreads();
    }
    if (threadIdx.x == 0) {
        const float ms    = s2[0];
        const float recon = s1[0] * TD / (ms * ms);
        out[0] = recon + s3[0] / (float)BN;   // ALPHA = 1
    }
}

// ---------------------------------------------------------------------------
extern "C" void kernel_launch(void* const* d_in, const int* in_sizes, int n_in,
                              void* d_out, int out_size, void* d_ws, size_t ws_size,
                              hipStream_t stream)
{
    const float* X     = (const float*)d_in[0];
    const float* msk   = (const float*)d_in[1];
    const float* Xr    = (const float*)d_in[2];
    const float* mu_c  = (const float*)d_in[3];
    const float* var_c = (const float*)d_in[4];
    const float* phi_c = (const float*)d_in[5];
    const float* z     = (const float*)d_in[6];
    const float* mu_t  = (const float*)d_in[7];
    const float* st_t  = (const float*)d_in[8];
    float* ws = (float*)d_ws;

    const int n4 = in_sizes[0] / 4;   // 33,554,432 / 4

    vader_prep  <<<KN,           256, 0, stream>>>(mu_c, var_c, phi_c, ws);
    vader_recon <<<RECON_BLOCKS, 256, 0, stream>>>((const v4f*)X,
                                                   (const v4f*)msk,
                                                   (const v4f*)Xr, ws, n4);
    vader_latent<<<LAT_BLOCKS,   128, 0, stream>>>(z, mu_t, st_t, ws);
    vader_final <<<1,            256, 0, stream>>>(ws, (float*)d_out);
}